// EmbeddingBackbone_69011534512380
// MI455X (gfx1250) — compile-verified
//
#include <hip/hip_runtime.h>
#include <hip/hip_bf16.h>
#include <math.h>

// ---------------------------------------------------------------------------
// MI455X (gfx1250) fused LN+GEMM / GNN message-passing pipeline.
// All matmuls run on the CDNA5 matrix pipe via V_WMMA_F32_16X16X4_F32.
// Each block: 32 rows x 128 cols; each wave32 owns two 16x16 D tiles
// (dual accumulators) so every B-operand pair feeds two WMMA issues.
// ---------------------------------------------------------------------------

typedef __attribute__((ext_vector_type(2))) float v2f;
typedef __attribute__((ext_vector_type(8))) float v8f;

#define HDIM 128
#define LN_EPS 1e-5f

__device__ __forceinline__ float wave_sum32(float v) {
  #pragma unroll
  for (int off = 16; off > 0; off >>= 1) v += __shfl_xor(v, off, 32);
  return v;
}

__device__ __forceinline__ v8f wmma4_f32(v2f a, v2f b, v8f c) {
  // D = A(16x4 f32) * B(4x16 f32) + C(16x16 f32), wave32
  return __builtin_amdgcn_wmma_f32_16x16x4_f32(
      false, a, false, b, (short)0, c, false, false);
}

// ---------------------------------------------------------------------------
// Kernel: optional LayerNorm of 128-wide rows, then Y = LN(X) @ W + bias.
// One block = 32 rows. g==nullptr -> skip LN (q_proj path).
// ---------------------------------------------------------------------------
__global__ __launch_bounds__(256) void ln_proj_kernel(
    const float* __restrict__ X, const float* __restrict__ g,
    const float* __restrict__ bvec, const float* __restrict__ W,
    const float* __restrict__ bias, float* __restrict__ Y, int nrows) {
  __shared__ float As[32][132];  // 132 % 64 == 4 -> conflict-free row access

  const int tid  = threadIdx.x;
  const int wave = tid >> 5;
  const int lane = tid & 31;
  const int base = blockIdx.x * 32;

  // Phase 1: each wave LayerNorms 4 rows (4 floats per lane).
  #pragma unroll
  for (int rr = 0; rr < 4; ++rr) {
    const int row = wave * 4 + rr;
    const int gr  = base + row;
    const int c   = lane * 4;
    float4 v = make_float4(0.f, 0.f, 0.f, 0.f);
    if (gr < nrows) v = *(const float4*)(X + (unsigned)gr * HDIM + c);
    if (g != nullptr) {
      float s  = wave_sum32(v.x + v.y + v.z + v.w);
      float s2 = wave_sum32(v.x * v.x + v.y * v.y + v.z * v.z + v.w * v.w);
      const float mu   = s * (1.f / HDIM);
      const float var  = s2 * (1.f / HDIM) - mu * mu;
      const float rstd = rsqrtf(var + LN_EPS);
      v.x = (v.x - mu) * rstd * g[c + 0] + bvec[c + 0];
      v.y = (v.y - mu) * rstd * g[c + 1] + bvec[c + 1];
      v.z = (v.z - mu) * rstd * g[c + 2] + bvec[c + 2];
      v.w = (v.w - mu) * rstd * g[c + 3] + bvec[c + 3];
    }
    *(float4*)&As[row][c] = v;
  }
  __syncthreads();

  // Phase 2: WMMA. wave -> 16-col tile; two 16-row tiles per wave.
  const int m  = lane & 15;
  const int kh = lane >> 4;
  const int nc = wave * 16 + m;
  v8f acc0 = {0.f, 0.f, 0.f, 0.f, 0.f, 0.f, 0.f, 0.f};
  v8f acc1 = {0.f, 0.f, 0.f, 0.f, 0.f, 0.f, 0.f, 0.f};
  #pragma unroll 4
  for (int k0 = 0; k0 < HDIM; k0 += 4) {
    const int kk = k0 + 2 * kh;
    v2f b;
    b.x = W[(unsigned)kk * HDIM + nc];
    b.y = W[(unsigned)(kk + 1) * HDIM + nc];
    const v2f a0 = *(const v2f*)&As[m][kk];
    const v2f a1 = *(const v2f*)&As[m + 16][kk];
    acc0 = wmma4_f32(a0, b, acc0);
    acc1 = wmma4_f32(a1, b, acc1);
  }

  // Phase 3: bias + store (uniform full-tile fast path).
  const float bv = bias[nc];
  if (base + 32 <= nrows) {
    #pragma unroll
    for (int r = 0; r < 8; ++r) {
      const int ro = base + r + 8 * kh;
      Y[(unsigned)ro * HDIM + nc] = acc0[r] + bv;
      Y[(unsigned)(ro + 16) * HDIM + nc] = acc1[r] + bv;
    }
  } else {
    #pragma unroll
    for (int r = 0; r < 8; ++r) {
      const int ro = base + r + 8 * kh;
      if (ro < nrows)      Y[(unsigned)ro * HDIM + nc] = acc0[r] + bv;
      if (ro + 16 < nrows) Y[(unsigned)(ro + 16) * HDIM + nc] = acc1[r] + bv;
    }
  }
}

// ---------------------------------------------------------------------------
// Degree kernels
// ---------------------------------------------------------------------------
__global__ __launch_bounds__(256) void deg_kernel(const int* __restrict__ tail,
                                                  float* __restrict__ deg, int E) {
  const int i = blockIdx.x * 256 + threadIdx.x;
  if (i < E) atomicAdd(&deg[tail[i]], 1.0f);
}

__global__ __launch_bounds__(256) void invdeg_kernel(float* __restrict__ deg, int N) {
  const int i = blockIdx.x * 256 + threadIdx.x;
  if (i < N) deg[i] = 1.0f / fmaxf(deg[i], 1.0f);
}

// ---------------------------------------------------------------------------
// Kernel: msg = (cur[head] + rel) @ msg_W + msg_b, scatter-add into agg[tail].
// One block = 32 edges.
// ---------------------------------------------------------------------------
__global__ __launch_bounds__(256) void msg_kernel(
    const float* __restrict__ cur, const float* __restrict__ rel,
    const int* __restrict__ head, const int* __restrict__ tail,
    const float* __restrict__ W, const float* __restrict__ bias,
    float* __restrict__ agg, int E) {
  __shared__ float As[32][132];
  __shared__ int tl[32];

  const int tid  = threadIdx.x;
  const int wave = tid >> 5;
  const int lane = tid & 31;
  const int base = blockIdx.x * 32;
  __shared__ int hd[32];

  if (tid < 32) {
    const int e = base + tid;
    const int h = (e < E) ? head[e] : 0;
    tl[tid] = (e < E) ? tail[e] : -1;
    hd[tid] = h;
    // CDNA5 prefetch of the randomly gathered source row (global_prefetch_b8)
    __builtin_prefetch(cur + (unsigned)h * HDIM, 0, 3);
  }
  __syncthreads();

  // Phase 1: stage A rows = cur[head[e]] + rel[e] (float4 LDS stores)
  {
    const int row = tid >> 3;
    const int c0  = (tid & 7) * 16;
    const int e   = base + row;
    if (e < E) {
      const float* rrow = rel + (unsigned)e * HDIM;
      const float* crow = cur + (unsigned)hd[row] * HDIM;
      #pragma unroll
      for (int j = 0; j < 4; ++j) {
        const int c = c0 + j * 4;
        float4 a = *(const float4*)(rrow + c);
        float4 bb = *(const float4*)(crow + c);
        a.x += bb.x; a.y += bb.y; a.z += bb.z; a.w += bb.w;
        *(float4*)&As[row][c] = a;
      }
    } else {
      const float4 z = make_float4(0.f, 0.f, 0.f, 0.f);
      #pragma unroll
      for (int j = 0; j < 4; ++j) *(float4*)&As[row][c0 + j * 4] = z;
    }
  }
  __syncthreads();

  // Phase 2: WMMA GEMM (two row tiles per wave)
  const int m  = lane & 15;
  const int kh = lane >> 4;
  const int nc = wave * 16 + m;
  v8f acc0 = {0.f, 0.f, 0.f, 0.f, 0.f, 0.f, 0.f, 0.f};
  v8f acc1 = {0.f, 0.f, 0.f, 0.f, 0.f, 0.f, 0.f, 0.f};
  #pragma unroll 4
  for (int k0 = 0; k0 < HDIM; k0 += 4) {
    const int kk = k0 + 2 * kh;
    v2f b;
    b.x = W[(unsigned)kk * HDIM + nc];
    b.y = W[(unsigned)(kk + 1) * HDIM + nc];
    const v2f a0 = *(const v2f*)&As[m][kk];
    const v2f a1 = *(const v2f*)&As[m + 16][kk];
    acc0 = wmma4_f32(a0, b, acc0);
    acc1 = wmma4_f32(a1, b, acc1);
  }

  // Phase 3: scatter-add into agg[tail[e]]
  const float bv = bias[nc];
  #pragma unroll
  for (int r = 0; r < 8; ++r) {
    const int mo = r + 8 * kh;
    const int t0 = tl[mo];
    const int t1 = tl[mo + 16];
    if (t0 >= 0) atomicAdd(&agg[(unsigned)t0 * HDIM + nc], acc0[r] + bv);
    if (t1 >= 0) atomicAdd(&agg[(unsigned)t1 * HDIM + nc], acc1[r] + bv);
  }
}

// ---------------------------------------------------------------------------
// Kernel: upd = [cur | agg*inv_deg] @ upd_W + upd_b  (K=256)
//         out = LN(cur + gelu(upd))
// One block = 32 node rows.
// ---------------------------------------------------------------------------
__global__ __launch_bounds__(256) void upd_kernel(
    const float* __restrict__ cur, const float* __restrict__ agg,
    const float* __restrict__ inv_deg, const float* __restrict__ W,
    const float* __restrict__ bias, const float* __restrict__ lng,
    const float* __restrict__ lnb, float* __restrict__ out, int N) {
  __shared__ float As[32][260];  // 260 % 64 == 4 -> conflict-free
  __shared__ float Ts[32][132];

  const int tid  = threadIdx.x;
  const int wave = tid >> 5;
  const int lane = tid & 31;
  const int base = blockIdx.x * 32;
  const bool full = (base + 32) <= N;

  // Phase 1: stage concat rows [cur(0:128) | agg*inv_deg(128:256)]
  {
    const int row = tid >> 3;
    const int c0  = (tid & 7) * 32;
    const int gr  = base + row;
    const float idg = (gr < N) ? inv_deg[gr] : 0.f;
    #pragma unroll
    for (int j = 0; j < 8; ++j) {
      const int c = c0 + j * 4;
      float4 v = make_float4(0.f, 0.f, 0.f, 0.f);
      if (gr < N) {
        if (c < HDIM) {
          v = *(const float4*)(cur + (unsigned)gr * HDIM + c);
        } else {
          v = *(const float4*)(agg + (unsigned)gr * HDIM + (c - HDIM));
          v.x *= idg; v.y *= idg; v.z *= idg; v.w *= idg;
        }
      }
      *(float4*)&As[row][c] = v;
    }
  }
  __syncthreads();

  // Phase 2: WMMA GEMM, K = 256
  const int m  = lane & 15;
  const int kh = lane >> 4;
  const int nc = wave * 16 + m;
  v8f acc0 = {0.f, 0.f, 0.f, 0.f, 0.f, 0.f, 0.f, 0.f};
  v8f acc1 = {0.f, 0.f, 0.f, 0.f, 0.f, 0.f, 0.f, 0.f};
  #pragma unroll 4
  for (int k0 = 0; k0 < 2 * HDIM; k0 += 4) {
    const int kk = k0 + 2 * kh;
    v2f b;
    b.x = W[(unsigned)kk * HDIM + nc];
    b.y = W[(unsigned)(kk + 1) * HDIM + nc];
    const v2f a0 = *(const v2f*)&As[m][kk];
    const v2f a1 = *(const v2f*)&As[m + 16][kk];
    acc0 = wmma4_f32(a0, b, acc0);
    acc1 = wmma4_f32(a1, b, acc1);
  }

  // Phase 3: residual + exact GELU, stage pre-LN rows
  const float bv = bias[nc];
  #pragma unroll
  for (int r = 0; r < 8; ++r) {
    const int mo = r + 8 * kh;
    const float u0 = acc0[r] + bv;
    const float u1 = acc1[r] + bv;
    const float g0 = 0.5f * u0 * (1.0f + erff(u0 * 0.70710678118654752f));
    const float g1 = 0.5f * u1 * (1.0f + erff(u1 * 0.70710678118654752f));
    Ts[mo][nc]      = As[mo][nc] + g0;       // As[:, 0:128] holds cur row
    Ts[mo + 16][nc] = As[mo + 16][nc] + g1;
  }
  __syncthreads();

  // Phase 4: LayerNorm (4 rows per wave) and write out
  #pragma unroll
  for (int rr = 0; rr < 4; ++rr) {
    const int row = wave * 4 + rr;
    const int gr  = base + row;
    const int c   = lane * 4;
    float4 v = *(const float4*)&Ts[row][c];
    float s  = wave_sum32(v.x + v.y + v.z + v.w);
    float s2 = wave_sum32(v.x * v.x + v.y * v.y + v.z * v.z + v.w * v.w);
    const float mu   = s * (1.f / HDIM);
    const float var  = s2 * (1.f / HDIM) - mu * mu;
    const float rstd = rsqrtf(var + LN_EPS);
    v.x = (v.x - mu) * rstd * lng[c + 0] + lnb[c + 0];
    v.y = (v.y - mu) * rstd * lng[c + 1] + lnb[c + 1];
    v.z = (v.z - mu) * rstd * lng[c + 2] + lnb[c + 2];
    v.w = (v.w - mu) * rstd * lng[c + 3] + lnb[c + 3];
    if (full || gr < N) *(float4*)(out + (unsigned)gr * HDIM + c) = v;
  }
}

// ---------------------------------------------------------------------------
// Launch
// ---------------------------------------------------------------------------
extern "C" void kernel_launch(void* const* d_in, const int* in_sizes, int n_in,
                              void* d_out, int out_size, void* d_ws, size_t ws_size,
                              hipStream_t stream) {
  const float* question_emb    = (const float*)d_in[0];
  const float* node_embeddings = (const float*)d_in[1];
  const float* edge_embeddings = (const float*)d_in[2];
  const float* node_norm_g     = (const float*)d_in[3];
  const float* node_norm_b     = (const float*)d_in[4];
  const float* rel_norm_g      = (const float*)d_in[5];
  const float* rel_norm_b      = (const float*)d_in[6];
  const float* node_proj_W     = (const float*)d_in[7];
  const float* node_proj_b     = (const float*)d_in[8];
  const float* rel_proj_W      = (const float*)d_in[9];
  const float* rel_proj_b      = (const float*)d_in[10];
  const float* q_proj_W        = (const float*)d_in[11];
  const float* q_proj_b        = (const float*)d_in[12];
  const float* msg_W           = (const float*)d_in[13];
  const float* msg_b           = (const float*)d_in[14];
  const float* upd_W           = (const float*)d_in[15];
  const float* upd_b           = (const float*)d_in[16];
  const float* ln_g            = (const float*)d_in[17];
  const float* ln_b            = (const float*)d_in[18];
  const int*   edge_index      = (const int*)d_in[19];

  const int Q = in_sizes[0] / HDIM;
  const int N = in_sizes[1] / HDIM;
  const int E = in_sizes[2] / HDIM;
  const int L = in_sizes[13] / (HDIM * HDIM);

  const int* head = edge_index;
  const int* tail = edge_index + E;

  float* out_final = (float*)d_out;                 // [N,128]
  float* rel_tok   = out_final + (size_t)N * HDIM;  // [E,128]
  float* q_tok     = rel_tok + (size_t)E * HDIM;    // [Q,128]

  float* ws    = (float*)d_ws;
  float* out_a = ws;                                // [N,128]
  float* out_b = out_a + (size_t)N * HDIM;          // [N,128]
  float* agg   = out_b + (size_t)N * HDIM;          // [N,128]
  float* deg   = agg + (size_t)N * HDIM;            // [N] (becomes inv_deg)

  const dim3 blk(256);

  // Projection stage
  ln_proj_kernel<<<(Q + 31) / 32, blk, 0, stream>>>(
      question_emb, nullptr, nullptr, q_proj_W, q_proj_b, q_tok, Q);
  ln_proj_kernel<<<(N + 31) / 32, blk, 0, stream>>>(
      node_embeddings, node_norm_g, node_norm_b, node_proj_W, node_proj_b,
      out_a, N);
  ln_proj_kernel<<<(E + 31) / 32, blk, 0, stream>>>(
      edge_embeddings, rel_norm_g, rel_norm_b, rel_proj_W, rel_proj_b,
      rel_tok, E);

  // Degrees
  hipMemsetAsync(deg, 0, (size_t)N * sizeof(float), stream);
  deg_kernel<<<(E + 255) / 256, blk, 0, stream>>>(tail, deg, E);
  invdeg_kernel<<<(N + 255) / 256, blk, 0, stream>>>(deg, N);

  // GNN layers
  const float* cur = out_a;
  for (int l = 0; l < L; ++l) {
    hipMemsetAsync(agg, 0, (size_t)N * HDIM * sizeof(float), stream);
    msg_kernel<<<(E + 31) / 32, blk, 0, stream>>>(
        cur, rel_tok, head, tail, msg_W + (size_t)l * HDIM * HDIM,
        msg_b + (size_t)l * HDIM, agg, E);
    float* dst = (l == L - 1) ? out_final : ((cur == out_a) ? out_b : out_a);
    upd_kernel<<<(N + 31) / 32, blk, 0, stream>>>(
        cur, agg, deg, upd_W + (size_t)l * 2 * HDIM * HDIM,
        upd_b + (size_t)l * HDIM, ln_g + (size_t)l * HDIM,
        ln_b + (size_t)l * HDIM, dst, N);
    cur = dst;
  }
  if (L == 0) {
    hipMemcpyAsync(out_final, out_a, (size_t)N * HDIM * sizeof(float),
                   hipMemcpyDeviceToDevice, stream);
  }
}